// UnpoolingLayer_23012434772655
// MI455X (gfx1250) — compile-verified
//
#include <hip/hip_runtime.h>
#include <hip/hip_bf16.h>
#include <cstddef>
#include <cstdint>

// Problem constants (from reference)
#define N_COARSE 2048
#define N_FINE   16384
#define D_COARSE 512
#define D_FINE   256
#define KC       32          // K-chunk staged in LDS
#define CG_ITERS 40

typedef __attribute__((ext_vector_type(2))) float v2f;
typedef __attribute__((ext_vector_type(8))) float v8f;
typedef __attribute__((ext_vector_type(4))) unsigned int u32x4;
typedef __attribute__((ext_vector_type(8))) int i32x8;
typedef __attribute__((ext_vector_type(4))) int i32x4;

// ---------------------------------------------------------------------------
// TDM descriptor build + issue: 2D tile (rows x KC floats) from row-major
// global memory (row stride = stride_elems floats) into LDS at lds_byte.
// LDS padding: pad_interval = 32 dwords (code 4), pad_amount = 2 dwords
// (code 1)  ->  each 32-float row lands on a 34-float stride in LDS.
// D# bitfields per CDNA5 ISA 8.3/8.4. Uses the 6-arg clang-23 builtin.
// ---------------------------------------------------------------------------
__device__ __forceinline__ void tdm_load_tile(uint32_t lds_byte,
                                              const float* gptr,
                                              uint32_t rows,
                                              uint32_t stride_elems) {
    const uint64_t ga = (uint64_t)(uintptr_t)gptr;
    u32x4 g0;
    g0[0] = 1u;                                   // count=1, user descriptor
    g0[1] = lds_byte;                             // lds_addr
    g0[2] = (uint32_t)ga;                         // global_addr[31:0]
    g0[3] = (uint32_t)((ga >> 32) & 0x01FFFFFFu)  // global_addr[56:32]
          | (2u << 30);                           // type = 2 ("image")
    i32x8 g1;
    g1[0] = (int)((2u << 16)      // data_size = 4 bytes
                | (1u << 20)      // pad_enable
                | (4u << 22)      // pad_interval: 32 dwords
                | (1u << 25));    // pad_amount:   2 dwords
    g1[1] = 0;                                    // barrier addr 0; dim0 lo16=0
    g1[2] = (int)0x4000;                          // dim0 hi16 (dim0=2^30); dim1 lo16=0
    g1[3] = (int)(0x4000u | ((uint32_t)KC << 16));// dim1 hi16; tile_dim0 = 32
    g1[4] = (int)rows;                            // tile_dim1 = rows, tile_dim2 = 0
    g1[5] = (int)stride_elems;                    // tensor_dim0_stride[31:0]
    g1[6] = 0;                                    // stride0 hi / stride1 lo
    g1[7] = 0;
    const i32x4 z4 = {0, 0, 0, 0};
    const i32x8 z8 = {0, 0, 0, 0, 0, 0, 0, 0};
    __builtin_amdgcn_tensor_load_to_lds(g0, g1, z4, z4, z8, 0);
}

// ---------------------------------------------------------------------------
// G = C * C^T   (2048 x 2048), C is 2048 x 16384 row-major.
// WG = 256 threads = 8 waves (2x4 grid); WG tile = 64(M) x 128(N); each wave
// computes a 32x32 block as 2x2 WMMA tiles (V_WMMA_F32_16X16X4_F32), so each
// LDS fragment is reused twice. Tiles are streamed by the Tensor Data Mover
// into double-buffered LDS (s_wait_tensorcnt + two barriers per chunk).
// Fragment reads use opaque inline-asm ds_load_b64 (early-clobber outputs,
// immediate DS offsets, "memory" clobber) because the TDM's LDS writes are
// invisible to the compiler's alias analysis.
// ---------------------------------------------------------------------------
#define LDS_STRIDE 34                      // 32 data + 2 pad floats per row
#define A_ROWS 64
#define B_ROWS 128
#define A_FLOATS (A_ROWS * LDS_STRIDE)     // 2176
#define B_FLOATS (B_ROWS * LDS_STRIDE)     // 4352
// Byte distance between the two fragments of a pair: 16 rows * 34 * 4 = 2176.

__global__ __launch_bounds__(256) void gram_kernel(const float* __restrict__ C,
                                                   float* __restrict__ G) {
    __shared__ __align__(16) float lds[2 * A_FLOATS + 2 * B_FLOATS];   // ~51 KB
    const int tid  = threadIdx.x;
    const int wave = tid >> 5;
    const int lane = tid & 31;
    const int lh   = lane >> 4;            // lane-half: K pair select / M+8
    const int ll   = lane & 15;
    const int i0 = blockIdx.y * 64;        // M rows of G (rows of C)
    const int j0 = blockIdx.x * 128;       // N cols of G (rows of C)
    const int m_base = (wave >> 2) * 32;   // 0 or 32
    const int n_base = (wave & 3) * 32;    // 0..96

    // Low 32 bits of the flat address of an LDS object = LDS byte offset.
    const uint32_t lds_base = (uint32_t)(uintptr_t)(void*)lds;

    v8f acc00 = {}, acc01 = {}, acc10 = {}, acc11 = {};

    // Prologue: DMA first chunk into buffer 0 (wave 0 issues, EXEC-independent)
    if (wave == 0) {
        tdm_load_tile(lds_base, C + (size_t)i0 * N_FINE, A_ROWS, N_FINE);
        tdm_load_tile(lds_base + (uint32_t)(2 * A_FLOATS) * 4u,
                      C + (size_t)j0 * N_FINE, B_ROWS, N_FINE);
    }

    // Per-wave fragment base byte offsets within a buffer
    const uint32_t aoff = (uint32_t)(((m_base + ll) * LDS_STRIDE + 2 * lh) * 4);
    const uint32_t boff = (uint32_t)((2 * A_FLOATS + (n_base + ll) * LDS_STRIDE + 2 * lh) * 4);

    const int NCHUNK = N_FINE / KC;
    for (int it = 0; it < NCHUNK; ++it) {
        const int cur = it & 1;
        if (wave == 0) __builtin_amdgcn_s_wait_tensorcnt(0);
        __syncthreads();   // publish buf[cur]; also: everyone done reading buf[1-cur]
        if (wave == 0 && it + 1 < NCHUNK) {
            const int k1 = (it + 1) * KC;
            const int nxt = cur ^ 1;
            tdm_load_tile(lds_base + (uint32_t)(nxt * A_FLOATS) * 4u,
                          C + (size_t)i0 * N_FINE + k1, A_ROWS, N_FINE);
            tdm_load_tile(lds_base + (uint32_t)(2 * A_FLOATS + nxt * B_FLOATS) * 4u,
                          C + (size_t)j0 * N_FINE + k1, B_ROWS, N_FINE);
        }
        const uint32_t abase = lds_base + (uint32_t)(cur * A_FLOATS) * 4u + aoff;
        const uint32_t bbase = lds_base + (uint32_t)(cur * B_FLOATS) * 4u + boff;

        // One K-step: 4 ds_load_b64 (immediate offsets, 2 base regs) + 4 WMMAs.
        // O1 = kk*4 ; O2 = kk*4 + 2176 (fragment pair distance).
#define GRAM_STEP(O1, O2)                                                     \
        do {                                                                  \
            v2f a0, a1, b0, b1;                                               \
            asm volatile(                                                     \
                "ds_load_b64 %0, %4 offset:" O1 "\n\t"                        \
                "ds_load_b64 %1, %4 offset:" O2 "\n\t"                        \
                "ds_load_b64 %2, %5 offset:" O1 "\n\t"                        \
                "ds_load_b64 %3, %5 offset:" O2 "\n\t"                        \
                "s_wait_dscnt 0x0"                                            \
                : "=&v"(a0), "=&v"(a1), "=&v"(b0), "=&v"(b1)                  \
                : "v"(abase), "v"(bbase)                                      \
                : "memory");                                                  \
            acc00 = __builtin_amdgcn_wmma_f32_16x16x4_f32(false, a0, false, b0, \
                                                          (short)0, acc00, false, false); \
            acc01 = __builtin_amdgcn_wmma_f32_16x16x4_f32(false, a0, false, b1, \
                                                          (short)0, acc01, false, false); \
            acc10 = __builtin_amdgcn_wmma_f32_16x16x4_f32(false, a1, false, b0, \
                                                          (short)0, acc10, false, false); \
            acc11 = __builtin_amdgcn_wmma_f32_16x16x4_f32(false, a1, false, b1, \
                                                          (short)0, acc11, false, false); \
        } while (0)

        GRAM_STEP("0",   "2176");
        GRAM_STEP("16",  "2192");
        GRAM_STEP("32",  "2208");
        GRAM_STEP("48",  "2224");
        GRAM_STEP("64",  "2240");
        GRAM_STEP("80",  "2256");
        GRAM_STEP("96",  "2272");
        GRAM_STEP("112", "2288");
#undef GRAM_STEP

        __syncthreads();   // all reads of buf[cur] done before it is re-filled
    }

    const int gr = i0 + m_base + lh * 8;
    const int gc = j0 + n_base + ll;
    #pragma unroll
    for (int v = 0; v < 8; ++v) {
        G[(size_t)(gr + v) * N_COARSE + gc]           = acc00[v];
        G[(size_t)(gr + v) * N_COARSE + gc + 16]      = acc01[v];
        G[(size_t)(gr + 16 + v) * N_COARSE + gc]      = acc10[v];
        G[(size_t)(gr + 16 + v) * N_COARSE + gc + 16] = acc11[v];
    }
}

// ---------------------------------------------------------------------------
// h_fine = h_coarse[assign] @ W^T + b  -> (16384 x 256)
// WG tile 32(M) x 64(N), K over 512 in chunks of 32; gathered A rows.
// ---------------------------------------------------------------------------
__global__ __launch_bounds__(256) void hfine_kernel(const float* __restrict__ H,
                                                    const int* __restrict__ assign,
                                                    const float* __restrict__ W,
                                                    const float* __restrict__ bias,
                                                    float* __restrict__ out) {
    __shared__ float At[KC][32];   // At[k][m] = H[assign[m0+m]][k0+k]
    __shared__ float Bt[KC][64];   // Bt[k][n] = W[n0+n][k0+k]
    __shared__ int   rowsrc[32];
    const int tid  = threadIdx.x;
    const int wave = tid >> 5;
    const int lane = tid & 31;
    const int lh   = lane >> 4;
    const int ll   = lane & 15;
    const int m0 = blockIdx.y * 32;
    const int n0 = blockIdx.x * 64;
    const int m_off = (wave >> 2) * 16;
    const int n_off = (wave & 3) * 16;

    if (tid < 32) rowsrc[tid] = assign[m0 + tid];

    v8f acc = {};

    for (int k0 = 0; k0 < D_COARSE; k0 += KC) {
        __syncthreads();   // also publishes rowsrc on first iteration
        {   // gathered A block
            const int r  = tid >> 3;
            const int c4 = (tid & 7) * 4;
            const int src = rowsrc[r];
            const float4 v = *(const float4*)(H + (size_t)src * D_COARSE + k0 + c4);
            At[c4 + 0][r] = v.x; At[c4 + 1][r] = v.y;
            At[c4 + 2][r] = v.z; At[c4 + 3][r] = v.w;
        }
        #pragma unroll
        for (int it = 0; it < 2; ++it) {  // B block from W (row-major 256x512)
            const int idx = tid + it * 256;
            const int r  = idx >> 3;
            const int c4 = (idx & 7) * 4;
            const float4 v = *(const float4*)(W + (size_t)(n0 + r) * D_COARSE + k0 + c4);
            Bt[c4 + 0][r] = v.x; Bt[c4 + 1][r] = v.y;
            Bt[c4 + 2][r] = v.z; Bt[c4 + 3][r] = v.w;
        }
        __syncthreads();
        #pragma unroll
        for (int kk = 0; kk < KC; kk += 4) {
            v2f a, b;
            a.x = At[kk + 2 * lh + 0][m_off + ll];
            a.y = At[kk + 2 * lh + 1][m_off + ll];
            b.x = Bt[kk + 2 * lh + 0][n_off + ll];
            b.y = Bt[kk + 2 * lh + 1][n_off + ll];
            acc = __builtin_amdgcn_wmma_f32_16x16x4_f32(false, a, false, b,
                                                        (short)0, acc, false, false);
        }
    }
    const int gc = n0 + n_off + ll;
    const int gr = m0 + m_off + lh * 8;
    const float bb = bias[gc];
    #pragma unroll
    for (int v = 0; v < 8; ++v)
        out[(size_t)(gr + v) * D_FINE + gc] = acc[v] + bb;
}

// ---------------------------------------------------------------------------
// CG solve (G y = x_coarse, 3 RHS). Vectors stored row-major (2048 x 3).
// scal layout: [0..2]=rsold, [3..5]=pAp, [6..8]=rsnew
// ---------------------------------------------------------------------------
__global__ __launch_bounds__(256) void cg_init(const float* __restrict__ x,
                                               float* __restrict__ r,
                                               float* __restrict__ p,
                                               float* __restrict__ y,
                                               float* __restrict__ scal) {
    const int i = blockIdx.x * 256 + threadIdx.x;
    if (i < N_COARSE * 3) { const float v = x[i]; r[i] = v; p[i] = v; y[i] = 0.f; }
    if (blockIdx.x == 0 && threadIdx.x < 16) scal[threadIdx.x] = 0.f;
}

__global__ __launch_bounds__(256) void cg_matvec(const float* __restrict__ G,
                                                 const float* __restrict__ p,
                                                 float* __restrict__ Ap) {
    const int wave = threadIdx.x >> 5, lane = threadIdx.x & 31;
    const int row = blockIdx.x * 8 + wave;
    const float* g = G + (size_t)row * N_COARSE;
    float a0 = 0.f, a1 = 0.f, a2 = 0.f;
    for (int c = lane; c < N_COARSE; c += 32) {
        const float gv = g[c];
        a0 = fmaf(gv, p[c * 3 + 0], a0);
        a1 = fmaf(gv, p[c * 3 + 1], a1);
        a2 = fmaf(gv, p[c * 3 + 2], a2);
    }
    #pragma unroll
    for (int off = 16; off > 0; off >>= 1) {   // wave32 butterfly reduce
        a0 += __shfl_xor(a0, off, 32);
        a1 += __shfl_xor(a1, off, 32);
        a2 += __shfl_xor(a2, off, 32);
    }
    if (lane == 0) {
        Ap[row * 3 + 0] = a0; Ap[row * 3 + 1] = a1; Ap[row * 3 + 2] = a2;
    }
}

__global__ __launch_bounds__(256) void dot3_kernel(const float* __restrict__ a,
                                                   const float* __restrict__ b,
                                                   float* __restrict__ out3) {
    __shared__ float sm[3 * 256];
    const int tid = threadIdx.x;
    float s0 = 0.f, s1 = 0.f, s2 = 0.f;
    for (int c = tid; c < N_COARSE; c += 256) {
        s0 = fmaf(a[c * 3 + 0], b[c * 3 + 0], s0);
        s1 = fmaf(a[c * 3 + 1], b[c * 3 + 1], s1);
        s2 = fmaf(a[c * 3 + 2], b[c * 3 + 2], s2);
    }
    sm[tid] = s0; sm[256 + tid] = s1; sm[512 + tid] = s2;
    __syncthreads();
    for (int s = 128; s > 0; s >>= 1) {
        if (tid < s) {
            sm[tid] += sm[tid + s];
            sm[256 + tid] += sm[256 + tid + s];
            sm[512 + tid] += sm[512 + tid + s];
        }
        __syncthreads();
    }
    if (tid == 0) { out3[0] = sm[0]; out3[1] = sm[256]; out3[2] = sm[512]; }
}

__global__ __launch_bounds__(256) void cg_update_xr(float* __restrict__ y,
                                                    float* __restrict__ r,
                                                    const float* __restrict__ p,
                                                    const float* __restrict__ Ap,
                                                    const float* __restrict__ scal) {
    const int i = blockIdx.x * 256 + threadIdx.x;
    if (i < N_COARSE * 3) {
        const int d = i % 3;
        const float den = scal[3 + d];
        const float alpha = scal[d] / (den == 0.f ? 1.f : den);
        y[i] = fmaf(alpha, p[i], y[i]);
        r[i] = fmaf(-alpha, Ap[i], r[i]);
    }
}

__global__ __launch_bounds__(256) void cg_update_p(float* __restrict__ p,
                                                   const float* __restrict__ r,
                                                   const float* __restrict__ scal) {
    const int i = blockIdx.x * 256 + threadIdx.x;
    if (i < N_COARSE * 3) {
        const int d = i % 3;
        const float den = scal[d];
        const float beta = scal[6 + d] / (den == 0.f ? 1.f : den);
        p[i] = fmaf(beta, p[i], r[i]);
    }
}

__global__ void cg_shift(float* __restrict__ scal) {
    if (threadIdx.x < 3) scal[threadIdx.x] = scal[6 + threadIdx.x];  // rsold = rsnew
}

// ---------------------------------------------------------------------------
// x_fine = C^T @ y : thread f accumulates column f of C against y (3 cols).
// ---------------------------------------------------------------------------
__global__ __launch_bounds__(256) void xfine_kernel(const float* __restrict__ C,
                                                    const float* __restrict__ y,
                                                    float* __restrict__ xf) {
    const int f = blockIdx.x * 256 + threadIdx.x;
    float a0 = 0.f, a1 = 0.f, a2 = 0.f;
    for (int c = 0; c < N_COARSE; ++c) {
        const float v = C[(size_t)c * N_FINE + f];
        a0 = fmaf(v, y[c * 3 + 0], a0);
        a1 = fmaf(v, y[c * 3 + 1], a1);
        a2 = fmaf(v, y[c * 3 + 2], a2);
    }
    xf[f * 3 + 0] = a0; xf[f * 3 + 1] = a1; xf[f * 3 + 2] = a2;
}

// ---------------------------------------------------------------------------
extern "C" void kernel_launch(void* const* d_in, const int* in_sizes, int n_in,
                              void* d_out, int out_size, void* d_ws, size_t ws_size,
                              hipStream_t stream) {
    const float* h_coarse = (const float*)d_in[0];   // 2048 x 512
    const float* x_coarse = (const float*)d_in[1];   // 2048 x 3
    const float* Cmat     = (const float*)d_in[2];   // 2048 x 16384
    const int*   assign   = (const int*)d_in[3];     // 16384
    const float* W        = (const float*)d_in[4];   // 256 x 512
    const float* bias     = (const float*)d_in[5];   // 256

    float* h_fine = (float*)d_out;                            // 16384 x 256
    float* x_fine = (float*)d_out + (size_t)N_FINE * D_FINE;  // 16384 x 3

    // Workspace layout (floats)
    float* ws   = (float*)d_ws;
    float* G    = ws;                                  // 2048*2048
    float* p    = G  + (size_t)N_COARSE * N_COARSE;    // 2048*3
    float* r    = p  + N_COARSE * 3;
    float* yv   = r  + N_COARSE * 3;
    float* Ap   = yv + N_COARSE * 3;
    float* scal = Ap + N_COARSE * 3;                   // 16 floats

    // 1) h_fine: gather + WMMA GEMM (independent of the solve chain)
    hfine_kernel<<<dim3(D_FINE / 64, N_FINE / 32), 256, 0, stream>>>(
        h_coarse, assign, W, bias, h_fine);

    // 2) G = C C^T via fp32 WMMA + TDM double-buffered LDS streaming
    gram_kernel<<<dim3(N_COARSE / 128, N_COARSE / 64), 256, 0, stream>>>(Cmat, G);

    // 3) CG solve: G y = x_coarse (kappa ~ 4.4 -> 40 iters >> machine eps)
    const int nvec_blocks = (N_COARSE * 3 + 255) / 256;
    cg_init<<<nvec_blocks, 256, 0, stream>>>(x_coarse, r, p, yv, scal);
    dot3_kernel<<<1, 256, 0, stream>>>(r, r, scal + 0);          // rsold
    for (int it = 0; it < CG_ITERS; ++it) {
        cg_matvec<<<N_COARSE / 8, 256, 0, stream>>>(G, p, Ap);
        dot3_kernel<<<1, 256, 0, stream>>>(p, Ap, scal + 3);     // pAp
        cg_update_xr<<<nvec_blocks, 256, 0, stream>>>(yv, r, p, Ap, scal);
        dot3_kernel<<<1, 256, 0, stream>>>(r, r, scal + 6);      // rsnew
        cg_update_p<<<nvec_blocks, 256, 0, stream>>>(p, r, scal);
        cg_shift<<<1, 32, 0, stream>>>(scal);
    }

    // 4) x_fine = C^T y
    xfine_kernel<<<N_FINE / 256, 256, 0, stream>>>(Cmat, yv, x_fine);
}